// IntrinsicLearner_32710470926494
// MI455X (gfx1250) — compile-verified
//
#include <hip/hip_runtime.h>
#include <math.h>

typedef __attribute__((ext_vector_type(2))) float v2f;
typedef __attribute__((ext_vector_type(8))) float v8f;

#define NCTX 32
#define RANK 64
#define DD   2048   /* ID / NCTX */
#define LL   8192   /* == DIM */

// ---------------------------------------------------------------------------
// Kernel 1: recon[l,d] = sum_r U[l,r]*s[r]*V[d,r] + shared_V[d]
// One wave (32 threads) per 16x16 output tile, f32 WMMA, K-loop of 16 steps.
// A-matrix (16x4 f32): lanes 0-15 hold K={k0,k0+1}, lanes 16-31 hold K={k0+2,k0+3}
// B-matrix (4x16 f32): mirrored striping (N = lane&15).
// C/D (16x16 f32, 8 VGPRs): row M = j + 8*(lane>>4), col N = lane&15.
// ---------------------------------------------------------------------------
__global__ __launch_bounds__(32)
void recon_wmma_kernel(const float* __restrict__ U,
                       const float* __restrict__ s,
                       const float* __restrict__ V,
                       const float* __restrict__ sharedV,
                       float* __restrict__ recon) {
    const int lane = threadIdx.x;        // 0..31, EXEC all-ones (no divergence)
    const int n0   = blockIdx.x * 16;    // 0..2047 in steps of 16
    const int m0   = blockIdx.y * 16;    // 0 or 16
    const int mi   = m0 + (lane & 15);   // A row this lane supplies
    const int nn   = n0 + (lane & 15);   // B column this lane supplies
    const int kb   = (lane >> 4) * 2;    // K sub-offset per half-wave

    v8f acc = {};
#pragma unroll
    for (int k0 = 0; k0 < RANK; k0 += 4) {
        const int kk = k0 + kb;
        v2f a  = *(const v2f*)(U + mi * RANK + kk);   // U[mi, kk..kk+1]
        v2f sv = *(const v2f*)(s + kk);               // s[kk..kk+1]
        a.x *= sv.x;
        a.y *= sv.y;                                  // A = U .* s
        v2f b  = *(const v2f*)(V + nn * RANK + kk);   // B[k,n] = V[n,k]
        acc = __builtin_amdgcn_wmma_f32_16x16x4_f32(
            /*neg_a=*/false, a, /*neg_b=*/false, b,
            /*c_mod=*/(short)0, acc, /*reuse_a=*/false, /*reuse_b=*/false);
    }

    const float svn  = sharedV[nn];            // bias for this output column
    const int   half = (lane >> 4) * 8;
    const int   col  = n0 + (lane & 15);
#pragma unroll
    for (int j = 0; j < 8; ++j) {
        const int row = m0 + half + j;
        recon[row * DD + col] = acc[j] + svn;
    }
}

// ---------------------------------------------------------------------------
// Kernel 2: per-row fastfood + epilogue. One 512-thread block per ctx row.
//   buf = pad(recon)*BB ; WHT ; gather(Pi)*GG ; WHT ; out = initial + buf/div
// LDS: 8192 floats (32 KB) transform buffer + 512-float reduction scratch.
// ---------------------------------------------------------------------------
__global__ __launch_bounds__(512)
void fastfood_kernel(const float* __restrict__ recon,
                     const float* __restrict__ BB,
                     const float* __restrict__ GG,
                     const int*   __restrict__ Pi,
                     const float* __restrict__ initial,
                     float*       __restrict__ out) {
    __shared__ float buf[LL];
    __shared__ float red[512];

    const int l   = blockIdx.x;
    const int tid = threadIdx.x;
    const float* bbr = BB + (size_t)l * LL;
    const float* ggr = GG + (size_t)l * LL;
    const int*   pir = Pi + (size_t)l * LL;
    const float* rcr = recon + (size_t)l * DD;

    // load + BB multiply (padding region is exactly zero)
#pragma unroll
    for (int e = 0; e < 16; ++e) {
        const int i = tid + e * 512;
        buf[i] = (i < DD) ? rcr[i] * bbr[i] : 0.0f;
    }
    __syncthreads();

    // WHT #1: 13 butterfly stages, 4096 butterflies each (8 per thread)
    for (int h = 1; h < LL; h <<= 1) {
#pragma unroll
        for (int e = 0; e < 8; ++e) {
            const int j = tid + e * 512;
            const int i = ((j & ~(h - 1)) << 1) | (j & (h - 1));
            const float a = buf[i];
            const float b = buf[i + h];
            buf[i]     = a + b;
            buf[i + h] = a - b;
        }
        __syncthreads();
    }

    // permutation gather staged in registers; accumulate GG^2 partial
    float vals[16];
    float part = 0.0f;
#pragma unroll
    for (int e = 0; e < 16; ++e) {
        const int i = tid + e * 512;
        vals[e] = buf[pir[i]];
        const float g = ggr[i];
        part += g * g;
    }
    red[tid] = part;
    __syncthreads();                       // all gathers complete before rewrite

#pragma unroll
    for (int e = 0; e < 16; ++e) {
        const int i = tid + e * 512;
        buf[i] = vals[e] * ggr[i];
    }
    // deterministic tree reduction of sum(GG^2)
    for (int sft = 256; sft > 0; sft >>= 1) {
        if (tid < sft) red[tid] += red[tid + sft];
        __syncthreads();
    }
    const float sumgg = red[0];

    // WHT #2
    for (int h = 1; h < LL; h <<= 1) {
#pragma unroll
        for (int e = 0; e < 8; ++e) {
            const int j = tid + e * 512;
            const int i = ((j & ~(h - 1)) << 1) | (j & (h - 1));
            const float a = buf[i];
            const float b = buf[i + h];
            buf[i]     = a + b;
            buf[i + h] = a - b;
        }
        __syncthreads();
    }

    // epilogue: divisor = sqrt(LL * sum(GG^2)) (sqrt(DIM/LL) == 1)
    const float inv = 1.0f / sqrtf((float)LL * sumgg);
    const float* ini = initial + (size_t)l * LL;
    float*       o   = out     + (size_t)l * LL;
#pragma unroll
    for (int e = 0; e < 16; ++e) {
        const int i = tid + e * 512;
        o[i] = ini[i] + buf[i] * inv;
    }
}

extern "C" void kernel_launch(void* const* d_in, const int* in_sizes, int n_in,
                              void* d_out, int out_size, void* d_ws, size_t ws_size,
                              hipStream_t stream) {
    const float* U       = (const float*)d_in[0];   // (32, 64)
    const float* s       = (const float*)d_in[1];   // (64,)
    const float* V       = (const float*)d_in[2];   // (2048, 64)
    const float* sharedV = (const float*)d_in[3];   // (2048,)
    const float* initial = (const float*)d_in[4];   // (32, 8192)
    const float* BB      = (const float*)d_in[5];   // (32, 8192)
    const float* GG      = (const float*)d_in[6];   // (32, 8192)
    const int*   Pi      = (const int*)  d_in[7];   // (32, 8192)
    float* out   = (float*)d_out;                   // (32, 8192)
    float* recon = (float*)d_ws;                    // 32*2048 floats = 256 KB

    (void)in_sizes; (void)n_in; (void)out_size; (void)ws_size;

    dim3 g1(DD / 16, NCTX / 16);                    // 128 x 2 tiles, 1 wave each
    recon_wmma_kernel<<<g1, 32, 0, stream>>>(U, s, V, sharedV, recon);
    fastfood_kernel<<<NCTX, 512, 0, stream>>>(recon, BB, GG, Pi, initial, out);
}